// LightGCN_61692910240182
// MI455X (gfx1250) — compile-verified
//
#include <hip/hip_runtime.h>

// ---------------------------------------------------------------------------
// LightGCN + factor-weighted scoring for MI455X (gfx1250, wave32).
//
// Phase 1 (propagation): memory/atomic bound (~6.5 GB traffic @ 23.3 TB/s).
//   -> coalesced fp32 loads + global_atomic_add_f32, one thread per (edge,dim).
// Phase 2 (scoring): fold 4x4 factor_weight into u, then score = <W^T u, i>
//   over 64 dims. 16 pairs/wave via V_WMMA_F32_16X16X4_F32, K-loop of 16,
//   scores read off the C-matrix diagonal (fp32 end to end).
// ---------------------------------------------------------------------------

typedef float v2f __attribute__((ext_vector_type(2)));
typedef float v8f __attribute__((ext_vector_type(8)));

#define NUSER  500000
#define NITEM  200000
#define NNODE  700000
#define DIM    64
#define NEDGE  2000000
#define NBATCH 131072

// --- init: X = concat(user_emb, item_emb); acc = X -------------------------
__global__ void lgcn_init(const float* __restrict__ ue, const float* __restrict__ ie,
                          float* __restrict__ x, float* __restrict__ acc) {
    size_t t = (size_t)blockIdx.x * blockDim.x + threadIdx.x;
    const size_t total = (size_t)NNODE * DIM;
    if (t >= total) return;
    const size_t usz = (size_t)NUSER * DIM;
    float v = (t < usz) ? ue[t] : ie[t - usz];
    x[t] = v;
    acc[t] = v;
}

// --- degree accumulation ---------------------------------------------------
__global__ void lgcn_degree(const int* __restrict__ eu, const int* __restrict__ ei,
                            float* __restrict__ deg) {
    int t = blockIdx.x * blockDim.x + threadIdx.x;
    if (t >= NEDGE) return;
    atomicAdd(&deg[eu[t]], 1.0f);
    atomicAdd(&deg[NUSER + ei[t]], 1.0f);
}

// --- deg -> 1/sqrt(deg) in place ------------------------------------------
__global__ void lgcn_invsqrt(float* __restrict__ deg) {
    int t = blockIdx.x * blockDim.x + threadIdx.x;
    if (t >= NNODE) return;
    float d = deg[t];
    deg[t] = (d > 0.0f) ? rsqrtf(d) : 0.0f;
}

// --- one propagation layer: xn += A * xc (symmetric, normalized) ----------
// thread t -> edge e = t/64, dim d = t%64; dim contiguous within a wave so
// edge index is wave-uniform (scalarized) and row accesses are coalesced.
__global__ void lgcn_propagate(const float* __restrict__ xc, float* __restrict__ xn,
                               const int* __restrict__ eu, const int* __restrict__ ei,
                               const float* __restrict__ inv) {
    size_t t = (size_t)blockIdx.x * blockDim.x + threadIdx.x;
    const size_t total = (size_t)NEDGE * DIM;
    if (t >= total) return;
    int e = (int)(t >> 6);
    int d = (int)(t & 63);
    int su = eu[e];
    int sv = NUSER + ei[e];
    float w = inv[su] * inv[sv];
    float mu = xc[(size_t)su * DIM + d] * w;   // user -> item message
    float mi = xc[(size_t)sv * DIM + d] * w;   // item -> user message
    atomicAdd(&xn[(size_t)sv * DIM + d], mu);
    atomicAdd(&xn[(size_t)su * DIM + d], mi);
}

// --- acc += xn -------------------------------------------------------------
__global__ void lgcn_accum(float* __restrict__ acc, const float* __restrict__ xn) {
    size_t t = (size_t)blockIdx.x * blockDim.x + threadIdx.x;
    if (t >= (size_t)NNODE * DIM) return;
    acc[t] += xn[t];
}

// --- scoring: 16 pairs per wave via V_WMMA_F32_16X16X4_F32 ----------------
// A = W-transformed user rows (16 x 64), B = item rows as 64 x 16 columns.
// C[m][m] = <W^T u_m, i_m>; score = C[m][m] / 16 (folds Z = acc/4 twice).
__global__ void __launch_bounds__(256)
lgcn_score(const float* __restrict__ Z, const float* __restrict__ W4,
           const int* __restrict__ users, const int* __restrict__ items,
           float* __restrict__ out) {
    __shared__ float lds[8 * 2048];          // 8 waves x (16x64 U' + 16x64 I)
    const int wave = threadIdx.x >> 5;
    const int lane = threadIdx.x & 31;
    float* U = &lds[wave * 2048];
    float* I = U + 1024;
    const int pairBase = blockIdx.x * 128 + wave * 16;

    float Wm[16];
#pragma unroll
    for (int j = 0; j < 16; ++j) Wm[j] = W4[j];   // uniform -> scalar loads

    // Stage U' = (W^T-mixed across factor blocks) user rows.
    for (int t = lane; t < 1024; t += 32) {
        int p = t >> 6, dim = t & 63;
        int g = dim >> 4, dd = dim & 15;
        size_t base = (size_t)users[pairBase + p] * DIM;
        float a = Wm[0 * 4 + g] * Z[base + 0 * 16 + dd]
                + Wm[1 * 4 + g] * Z[base + 1 * 16 + dd]
                + Wm[2 * 4 + g] * Z[base + 2 * 16 + dd]
                + Wm[3 * 4 + g] * Z[base + 3 * 16 + dd];
        U[t] = a;
    }
    // Stage item rows.
    for (int t = lane; t < 1024; t += 32) {
        int p = t >> 6, dim = t & 63;
        size_t base = (size_t)(NUSER + items[pairBase + p]) * DIM;
        I[t] = Z[base + dim];
    }
    __syncthreads();

    // ISA A layout (16x4 f32): lanes 0-15 M=0..15 carry {K0,K1}, lanes 16-31
    // carry {K2,K3}. B (4x16) mirrors it: lanes 0-15 N=0..15 rows {K0,K1},
    // lanes 16-31 rows {K2,K3}.
    v8f c = {};
    const int row = lane & 15;
    const int ksel = (lane >> 4) << 1;       // 0 for lanes 0-15, 2 for 16-31
#pragma unroll
    for (int k = 0; k < 16; ++k) {
        int kb = k * 4 + ksel;
        v2f a, b;
        a.x = U[row * 64 + kb];
        a.y = U[row * 64 + kb + 1];
        b.x = I[row * 64 + kb];
        b.y = I[row * 64 + kb + 1];
        c = __builtin_amdgcn_wmma_f32_16x16x4_f32(
                /*neg_a=*/false, a, /*neg_b=*/false, b,
                /*c_mod=*/(short)0, c, /*reuse_a=*/false, /*reuse_b=*/false);
    }

    // Diagonal extraction per the 16x16 f32 C/D layout:
    //   m < 8  : lane m,      VGPR m
    //   m >= 8 : lane m + 16, VGPR m - 8
    float cr[8];
#pragma unroll
    for (int j = 0; j < 8; ++j) cr[j] = c[j];
    if (lane < 8) {
        out[pairBase + lane] = cr[lane] * 0.0625f;
    } else if (lane >= 24) {
        out[pairBase + lane - 16] = cr[lane - 24] * 0.0625f;
    }
}

extern "C" void kernel_launch(void* const* d_in, const int* in_sizes, int n_in,
                              void* d_out, int out_size, void* d_ws, size_t ws_size,
                              hipStream_t stream) {
    const float* ue    = (const float*)d_in[0];
    const float* ie    = (const float*)d_in[1];
    const float* fw    = (const float*)d_in[2];
    const int*   eu    = (const int*)d_in[3];
    const int*   ei    = (const int*)d_in[4];
    const int*   users = (const int*)d_in[5];
    const int*   items = (const int*)d_in[6];
    float*       out   = (float*)d_out;

    const size_t nd = (size_t)NNODE * DIM;   // 44.8M floats per buffer
    float* bufA = (float*)d_ws;              // layer ping
    float* bufB = bufA + nd;                 // layer pong
    float* accb = bufB + nd;                 // running layer sum
    float* deg  = accb + nd;                 // degree -> inv-sqrt

    hipMemsetAsync(deg, 0, (size_t)NNODE * sizeof(float), stream);
    lgcn_init<<<(unsigned)((nd + 255) / 256), 256, 0, stream>>>(ue, ie, bufA, accb);
    lgcn_degree<<<(NEDGE + 255) / 256, 256, 0, stream>>>(eu, ei, deg);
    lgcn_invsqrt<<<(NNODE + 255) / 256, 256, 0, stream>>>(deg);

    float* xc = bufA;
    float* xn = bufB;
    for (int l = 0; l < 3; ++l) {
        hipMemsetAsync(xn, 0, nd * sizeof(float), stream);
        lgcn_propagate<<<(unsigned)(((size_t)NEDGE * DIM) / 256), 256, 0, stream>>>(
            xc, xn, eu, ei, deg);
        lgcn_accum<<<(unsigned)((nd + 255) / 256), 256, 0, stream>>>(accb, xn);
        float* tmp = xc; xc = xn; xn = tmp;
    }

    lgcn_score<<<NBATCH / 128, 256, 0, stream>>>(accb, fw, users, items, out);
}